// GCN_Layer_64845416235580
// MI455X (gfx1250) — compile-verified
//
#include <hip/hip_runtime.h>

typedef __attribute__((ext_vector_type(2))) float v2f;
typedef __attribute__((ext_vector_type(8))) float v8f;

#define DIN  64
#define DOUT 64

// ---------------------------------------------------------------------------
// Kernel A: y = x @ W^T using V_WMMA_F32_16X16X4_F32 (full f32 precision).
//   y[n][o] = sum_i x[n][i] * W[o][i]
// One wave32 computes a 16-node x 64-output tile: 16 K-steps x 4 N-tiles.
// A-frag (16x4 f32): lane m=lane&15 holds row M=m; half=lane>>4 selects
//   K pair {k0,k0+1} (half=0) or {k0+2,k0+3} (half=1); VGPR0/1 = within pair.
// B-frag (4x16 f32): lane m holds column N=m, same half/K-pair rule.
// C/D (16x16 f32): component v -> row v + 8*half, col = m.
// ---------------------------------------------------------------------------
__global__ __launch_bounds__(256) void gcn_xwt_wmma(
    const float* __restrict__ x, const float* __restrict__ W,
    float* __restrict__ y, int n_nodes)
{
    __shared__ float sW[DOUT * DIN];            // 16 KB of 320 KB WGP LDS
    for (int i = threadIdx.x; i < DOUT * DIN; i += 256) sW[i] = W[i];
    __syncthreads();

    const int wave = threadIdx.x >> 5;          // 8 waves / block
    const int lane = threadIdx.x & 31;
    const int m    = lane & 15;
    const int half = lane >> 4;
    const int tile = blockIdx.x * 8 + wave;     // 16 nodes per tile
    const int row0 = tile * 16;
    if (row0 >= n_nodes) return;                // wave-uniform: EXEC stays all-1s

    const int arow = min(row0 + m, n_nodes - 1);
    const float* __restrict__ xr = x + (size_t)arow * DIN;

    v8f acc[4] = {};                            // 4 N-tiles of 16 cols each

    #pragma unroll
    for (int k0 = 0; k0 < DIN; k0 += 4) {
        const int ka = k0 + 2 * half;
        v2f a;
        a.x = xr[ka];
        a.y = xr[ka + 1];
        #pragma unroll
        for (int nt = 0; nt < 4; ++nt) {
            const int n = nt * 16 + m;
            v2f bw;
            bw.x = sW[n * DIN + ka];
            bw.y = sW[n * DIN + ka + 1];
            acc[nt] = __builtin_amdgcn_wmma_f32_16x16x4_f32(
                /*neg_a=*/false, a, /*neg_b=*/false, bw,
                /*c_mod=*/(short)0, acc[nt],
                /*reuse_a=*/false, /*reuse_b=*/false);
        }
    }

    #pragma unroll
    for (int nt = 0; nt < 4; ++nt) {
        #pragma unroll
        for (int v = 0; v < 8; ++v) {
            const int r = row0 + v + 8 * half;
            if (r < n_nodes)
                y[(size_t)r * DOUT + nt * 16 + m] = acc[nt][v];
        }
    }
}

// ---------------------------------------------------------------------------
// Kernel B: out[n][c] = b[c]  (bias broadcast; scatter then accumulates on top)
// ---------------------------------------------------------------------------
__global__ __launch_bounds__(256) void gcn_bias_init(
    float* __restrict__ out, const float* __restrict__ b, int n_nodes)
{
    const int total = n_nodes * DOUT;
    for (int i = blockIdx.x * blockDim.x + threadIdx.x; i < total;
         i += gridDim.x * blockDim.x)
        out[i] = b[i & (DOUT - 1)];
}

// ---------------------------------------------------------------------------
// Kernel C: out[row[e]] += val[e] * y[col[e]]   (16 lanes/edge, float4 each)
// Working set is L2-resident; 4 global_atomic_add_f32 per lane.
// ---------------------------------------------------------------------------
__global__ __launch_bounds__(256) void gcn_edge_scatter(
    const int* __restrict__ erow, const int* __restrict__ ecol,
    const float* __restrict__ eval, const float* __restrict__ y,
    float* out, int n_edges)
{
    const int gid = blockIdx.x * 256 + threadIdx.x;
    const int e = gid >> 4;
    const int q = gid & 15;
    if (e >= n_edges) return;

    const int r   = erow[e];
    const int c   = ecol[e];
    const float v = eval[e];

    const float4* __restrict__ ys =
        (const float4*)(y + (size_t)c * DOUT);
    const float4 mv = ys[q];

    float* o = out + (size_t)r * DOUT + q * 4;
    atomicAdd(o + 0, v * mv.x);
    atomicAdd(o + 1, v * mv.y);
    atomicAdd(o + 2, v * mv.z);
    atomicAdd(o + 3, v * mv.w);
}

// ---------------------------------------------------------------------------
extern "C" void kernel_launch(void* const* d_in, const int* in_sizes, int n_in,
                              void* d_out, int out_size, void* d_ws, size_t ws_size,
                              hipStream_t stream)
{
    const float* x    = (const float*)d_in[0];
    const int*   erow = (const int*)  d_in[1];
    const int*   ecol = (const int*)  d_in[2];
    const float* ev   = (const float*)d_in[3];
    const float* W    = (const float*)d_in[4];
    const float* b    = (const float*)d_in[5];
    float* out = (float*)d_out;
    float* y   = (float*)d_ws;                  // [n_nodes, 64] scratch

    const int n_nodes = in_sizes[0] / DIN;
    const int n_edges = in_sizes[3];

    // A: y = x @ W^T   (WMMA)
    const int tiles = (n_nodes + 15) / 16;
    gcn_xwt_wmma<<<(tiles + 7) / 8, 256, 0, stream>>>(x, W, y, n_nodes);

    // B: out = broadcast(b)
    const int total = n_nodes * DOUT;
    int bgrid = (total + 255) / 256;
    if (bgrid > 4096) bgrid = 4096;
    gcn_bias_init<<<bgrid, 256, 0, stream>>>(out, b, n_nodes);

    // C: scatter-accumulate edges into out
    const long long sthreads = (long long)n_edges * 16;
    gcn_edge_scatter<<<(int)((sthreads + 255) / 256), 256, 0, stream>>>(
        erow, ecol, ev, y, out, n_edges);
}